// LocationExpertRouter_53446573032180
// MI455X (gfx1250) — compile-verified
//
#include <hip/hip_runtime.h>
#include <hip/hip_bf16.h>

// Problem constants (match reference)
#define B_ROWS 2048
#define D_DIM  512
#define V_DIM  8192
#define E_NUM  8

// Tiling
#define MT   16            // rows per M-tile (WMMA M)
#define KT   32            // K staged through LDS per iteration
#define LDP  (KT + 4)      // 36-float row stride: 16B-aligned + bank-conflict-free
#define NT   256           // N columns per block
#define NPW  64            // N columns per wave (4 subtiles of 16)
#define KSTAGES (D_DIM / KT)              // 16
#define MAX_TILES (B_ROWS / MT + E_NUM)   // 136 worst-case padded M-tiles

typedef float v2f __attribute__((ext_vector_type(2)));
typedef float v8f __attribute__((ext_vector_type(8)));

// Workspace layout (ints)
#define WS_COUNTS 0
#define WS_TPREF  8
#define WS_ROWMAP 32
#define ROWMAP_LEN (B_ROWS + E_NUM * MT)

// ---- CDNA5 async global->LDS path (guarded; falls back to sync staging) ----
#if defined(__has_builtin)
#if __has_builtin(__builtin_amdgcn_global_load_async_to_lds_b128) && \
    __has_builtin(__builtin_amdgcn_s_wait_asynccnt)
#define USE_ASYNC_LDS 1
#endif
#endif
#ifndef USE_ASYNC_LDS
#define USE_ASYNC_LDS 0
#endif

#if USE_ASYNC_LDS
// Param 0 expects: int __vector(4) addrspace(1)*  (prints as "__device__" in HIP)
// Param 1 expects: int __vector(4) addrspace(3)*  (LDS)
typedef int v4i_ __attribute__((vector_size(16)));
typedef __attribute__((address_space(1))) v4i_* gbl4_t;
typedef __attribute__((address_space(3))) v4i_* lds4_t;

__device__ __forceinline__ void async_copy16(const float* gsrc, float* ldst) {
    __builtin_amdgcn_global_load_async_to_lds_b128(
        (gbl4_t)(__attribute__((address_space(1))) void*)(void*)gsrc,
        (lds4_t)(__attribute__((address_space(3))) void*)(void*)ldst,
        0, 0);
}
#endif

__global__ void __launch_bounds__(256)
router_setup(const int* __restrict__ addrs, int* __restrict__ ws) {
    __shared__ int scnt[E_NUM];
    __shared__ int scur[E_NUM];
    const int tid = threadIdx.x;
    if (tid < E_NUM) scnt[tid] = 0;
    __syncthreads();

    for (int b = tid; b < B_ROWS; b += 256) {
        int e = addrs[b] & (E_NUM - 1);
        atomicAdd(&scnt[e], 1);
    }
    __syncthreads();

    if (tid == 0) {
        int tp = 0;
        ws[WS_TPREF + 0] = 0;
        for (int e = 0; e < E_NUM; ++e) {
            int c = scnt[e];
            ws[WS_COUNTS + e] = c;
            scur[e] = tp * MT;                 // padded row start for expert e
            tp += (c + MT - 1) / MT;
            ws[WS_TPREF + e + 1] = tp;
        }
    }
    __syncthreads();

    // Padding slots must hold a valid row index (their results are discarded)
    for (int i = tid; i < ROWMAP_LEN; i += 256) ws[WS_ROWMAP + i] = 0;
    __syncthreads();

    for (int b = tid; b < B_ROWS; b += 256) {
        int e = addrs[b] & (E_NUM - 1);
        int pos = atomicAdd(&scur[e], 1);
        ws[WS_ROWMAP + pos] = b;
    }
}

__global__ void __launch_bounds__(128)
moe_gemm(const float* __restrict__ x,
         const float* __restrict__ W,
         const float* __restrict__ bias,
         const int*  __restrict__ ws,
         float* __restrict__ out) {
    const int t = blockIdx.y;                       // padded M-tile id
    const int totalTiles = ws[WS_TPREF + E_NUM];
    if (t >= totalTiles) return;                    // uniform per block

    int e = 0;
    while (t >= ws[WS_TPREF + e + 1]) ++e;
    const int localbase = (t - ws[WS_TPREF + e]) * MT;
    const int cnt = ws[WS_COUNTS + e];
    const int rowbase = t * MT;
    const int v0 = blockIdx.x * NT;

#if USE_ASYNC_LDS
    __shared__ float As[2][MT][LDP];
    __shared__ float Bs[2][NT][LDP];
#else
    __shared__ float As[1][MT][LDP];
    __shared__ float Bs[1][NT][LDP];
#endif
    __shared__ int rows_s[MT];

    const int tid = threadIdx.x;
    if (tid < MT) rows_s[tid] = ws[WS_ROWMAP + rowbase + tid];
    __syncthreads();

    const int lane = tid & 31;
    const int wv   = tid >> 5;

    v8f acc[4] = {v8f{}, v8f{}, v8f{}, v8f{}};

    const float* __restrict__ Wexp = W + (size_t)e * V_DIM * D_DIM;

    // Staging geometry: thread -> (row = tid>>3, float4 slot = tid&7).
    // B tile: 256 rows, thread covers rows mrow, mrow+16, ..., mrow+240.
    const int mrow = tid >> 3;
    const int f4   = tid & 7;
    const float* aSrc = x + (size_t)rows_s[mrow] * D_DIM + f4 * 4;
    const float* bSrc = Wexp + (size_t)(v0 + mrow) * D_DIM + f4 * 4;

    const int ml    = lane & 15;
    const int khalf = (lane >> 4) << 1;

#if USE_ASYNC_LDS
    // ---- double-buffered async pipeline: 17 async copies per stage per wave ----
    {
        // prologue: stage 0 into buffer 0
        async_copy16(aSrc, &As[0][mrow][f4 * 4]);
        #pragma unroll
        for (int i = 0; i < 16; ++i)
            async_copy16(bSrc + (size_t)i * 16 * D_DIM, &Bs[0][mrow + i * 16][f4 * 4]);
    }
    for (int s = 0; s < KSTAGES; ++s) {
        const int cur = s & 1;
        if (s + 1 < KSTAGES) {
            const int k0n = (s + 1) * KT;
            async_copy16(aSrc + k0n, &As[cur ^ 1][mrow][f4 * 4]);
            #pragma unroll
            for (int i = 0; i < 16; ++i)
                async_copy16(bSrc + k0n + (size_t)i * 16 * D_DIM,
                             &Bs[cur ^ 1][mrow + i * 16][f4 * 4]);
            __builtin_amdgcn_s_wait_asynccnt(17);  // stage s complete (in-order)
        } else {
            __builtin_amdgcn_s_wait_asynccnt(0);
        }
        __syncthreads();

        const float (*Ac)[LDP] = As[cur];
        const float (*Bc)[LDP] = Bs[cur];
        #pragma unroll
        for (int h = 0; h < 2; ++h) {              // two batches of 4 k-steps
            v2f af[4];
            v2f bf[4][4];
            #pragma unroll
            for (int q = 0; q < 4; ++q) {
                const int kb = h * 16 + q * 4 + khalf;
                af[q].x = Ac[ml][kb];
                af[q].y = Ac[ml][kb + 1];
                #pragma unroll
                for (int ns = 0; ns < 4; ++ns) {
                    const int n = wv * NPW + ns * 16 + ml;
                    bf[q][ns].x = Bc[n][kb];
                    bf[q][ns].y = Bc[n][kb + 1];
                }
            }
            #pragma unroll
            for (int q = 0; q < 4; ++q) {
                #pragma unroll
                for (int ns = 0; ns < 4; ++ns) {
                    acc[ns] = __builtin_amdgcn_wmma_f32_16x16x4_f32(
                        false, af[q], false, bf[q][ns], (short)0, acc[ns],
                        false, false);
                }
            }
        }
        __syncthreads();   // protect buffer 'cur' before stage s+2 overwrites it
    }
#else
    // ---- fallback: synchronous staging (known-good) ----
    for (int k0 = 0; k0 < D_DIM; k0 += KT) {
        {
            const float4 av = *(const float4*)(aSrc + k0);
            As[0][mrow][f4 * 4 + 0] = av.x; As[0][mrow][f4 * 4 + 1] = av.y;
            As[0][mrow][f4 * 4 + 2] = av.z; As[0][mrow][f4 * 4 + 3] = av.w;
        }
        #pragma unroll
        for (int i = 0; i < 16; ++i) {
            const float4 bv = *(const float4*)(bSrc + k0 + (size_t)i * 16 * D_DIM);
            float* d = &Bs[0][mrow + i * 16][f4 * 4];
            d[0] = bv.x; d[1] = bv.y; d[2] = bv.z; d[3] = bv.w;
        }
        __syncthreads();

        #pragma unroll
        for (int h = 0; h < 2; ++h) {
            v2f af[4];
            v2f bf[4][4];
            #pragma unroll
            for (int q = 0; q < 4; ++q) {
                const int kb = h * 16 + q * 4 + khalf;
                af[q].x = As[0][ml][kb];
                af[q].y = As[0][ml][kb + 1];
                #pragma unroll
                for (int ns = 0; ns < 4; ++ns) {
                    const int n = wv * NPW + ns * 16 + ml;
                    bf[q][ns].x = Bs[0][n][kb];
                    bf[q][ns].y = Bs[0][n][kb + 1];
                }
            }
            #pragma unroll
            for (int q = 0; q < 4; ++q) {
                #pragma unroll
                for (int ns = 0; ns < 4; ++ns) {
                    acc[ns] = __builtin_amdgcn_wmma_f32_16x16x4_f32(
                        false, af[q], false, bf[q][ns], (short)0, acc[ns],
                        false, false);
                }
            }
        }
        __syncthreads();
    }
#endif

    // Epilogue: D 16x16 layout -> lanes 0-15: N=lane, M=r; lanes 16-31: M=r+8
    const int nl   = lane & 15;
    const int madd = (lane >> 4) * 8;
    #pragma unroll
    for (int ns = 0; ns < 4; ++ns) {
        const int n = v0 + wv * NPW + ns * 16 + nl;
        const float bv = bias[(size_t)e * V_DIM + n];
        #pragma unroll
        for (int r = 0; r < 8; ++r) {
            const int m = r + madd;
            if (localbase + m < cnt) {      // skip padded rows
                out[(size_t)rows_s[m] * V_DIM + n] = acc[ns][r] + bv;
            }
        }
    }
}

extern "C" void kernel_launch(void* const* d_in, const int* in_sizes, int n_in,
                              void* d_out, int out_size, void* d_ws, size_t ws_size,
                              hipStream_t stream) {
    const float* x     = (const float*)d_in[0];
    const int*   addrs = (const int*)d_in[1];   // pointer_addresses
    const float* W     = (const float*)d_in[2]; // [E, V, D]
    const float* bias  = (const float*)d_in[3]; // [E, V]
    float*       out   = (float*)d_out;         // [B, V]
    int*         ws    = (int*)d_ws;

    router_setup<<<1, 256, 0, stream>>>(addrs, ws);

    dim3 grid(V_DIM / NT, MAX_TILES);
    moe_gemm<<<grid, 128, 0, stream>>>(x, W, bias, ws, out);
}